// EdgeConvBlock_71880572666453
// MI455X (gfx1250) — compile-verified
//
#include <hip/hip_runtime.h>

#define BB   16
#define NN   2048
#define KK   16
#define FIN  64
#define LL   64
#define EPSBN 1e-5f

typedef __attribute__((ext_vector_type(2))) float v2f;
typedef __attribute__((ext_vector_type(8))) float v8f;

// Load a 16x64 f32 block as sixteen 16x4 WMMA A/B fragments.
// Documented 32-bit 16x4 layout: lanes 0-15 hold K=(4kc+0,4kc+1) in (x,y),
// lanes 16-31 hold K=(4kc+2,4kc+3).  node = per-lane row/col id.
__device__ __forceinline__ void load_x_frag(const float* __restrict__ x,
                                            int node, int koff, v2f* f) {
  const float2* p = (const float2*)(x + (size_t)node * FIN);
#pragma unroll
  for (int kc = 0; kc < 16; ++kc) {
    float2 t = p[2 * kc + (koff >> 1)];
    f[kc].x = t.x;
    f[kc].y = t.y;
  }
}

// ---------------------------------------------------------------------------
// Kernel 1: per-node squared norms + factorized edge-MLP projections
//   p1 = x @ (Wtop - Wbot),  p2 = x @ Wbot     (bias cancels in BatchNorm)
// One wave32 per 16 rows; WMMA f32 16x16x4, K=64 -> 16 chunks.
// ---------------------------------------------------------------------------
__global__ void __launch_bounds__(32)
proj_kernel(const float* __restrict__ x, const float* __restrict__ W,
            float* __restrict__ sq, float* __restrict__ p1,
            float* __restrict__ p2) {
  const int lane = threadIdx.x;
  const int half = lane >> 4;
  const int koff = half * 2;
  const int m16  = lane & 15;
  const int rowBase = blockIdx.x * 16;

  v2f a[16];
  load_x_frag(x, rowBase + m16, koff, a);

  // squared norm of each row (partial per lane, add the two half-wave parts)
  float s = 0.f;
#pragma unroll
  for (int kc = 0; kc < 16; ++kc) s += a[kc].x * a[kc].x + a[kc].y * a[kc].y;
  s += __shfl_xor(s, 16, 32);
  if (half == 0) sq[rowBase + m16] = s;

#pragma unroll
  for (int ct = 0; ct < 4; ++ct) {
    v8f c1 = {};
    v8f c2 = {};
#pragma unroll
    for (int kc = 0; kc < 16; ++kc) {
      const int k0 = 4 * kc + koff;
      const int cc = ct * 16 + m16;
      const float wt0 = W[(size_t)k0 * LL + cc];
      const float wt1 = W[(size_t)(k0 + 1) * LL + cc];
      const float wb0 = W[(size_t)(k0 + 64) * LL + cc];
      const float wb1 = W[(size_t)(k0 + 65) * LL + cc];
      v2f b1; b1.x = wt0 - wb0; b1.y = wt1 - wb1;
      v2f b2; b2.x = wb0;       b2.y = wb1;
      c1 = __builtin_amdgcn_wmma_f32_16x16x4_f32(false, a[kc], false, b1,
                                                 (short)0, c1, false, false);
      c2 = __builtin_amdgcn_wmma_f32_16x16x4_f32(false, a[kc], false, b2,
                                                 (short)0, c2, false, false);
    }
#pragma unroll
    for (int r = 0; r < 8; ++r) {
      const int mrow = rowBase + r + half * 8;   // C layout: M = r + 8*(lane/16)
      const int ccol = ct * 16 + m16;            //           N = lane % 16
      p1[(size_t)mrow * LL + ccol] = c1[r];
      p2[(size_t)mrow * LL + ccol] = c2[r];
    }
  }
}

// ---------------------------------------------------------------------------
// Kernel 2: fused Gram-matrix + top-K=16 selection (no NxN tensor in memory).
// One wave32 per 32 query rows of one event (two 16-row WMMA blocks share the
// same column-tile loads -> halves L2 traffic per FLOP).  128 column tiles.
// d[m][n] = sq[m] + sq[n] - 2 * dot(x_m, x_n)  computed via WMMA f32.
// Selection: lane l owns row l (32 rows / 32 lanes) -> no merge phase.
// ---------------------------------------------------------------------------
__global__ void __launch_bounds__(32)
knn_kernel(const float* __restrict__ x, const float* __restrict__ sq,
           int* __restrict__ idx) {
  __shared__ float tile[32][17];

  const int lane = threadIdx.x;
  const int half = lane >> 4;
  const int koff = half * 2;
  const int m16  = lane & 15;

  const int wg = blockIdx.x;
  const int b  = wg >> 6;                 // event        (NN/32 = 64 blocks)
  const int rt = wg & 63;                 // 32-row tile within event
  const int rowBase = b * NN + rt * 32;   // global node id of first row

  v2f a0[16];                             // rows rowBase+0  .. +15
  v2f a1[16];                             // rows rowBase+16 .. +31
  load_x_frag(x, rowBase + m16,      koff, a0);
  load_x_frag(x, rowBase + 16 + m16, koff, a1);
  const float sqr_self = sq[rowBase + lane];   // lane l owns row l

  float bd[16];
  int   bi[16];
#pragma unroll
  for (int j = 0; j < 16; ++j) { bd[j] = 3.0e38f; bi[j] = 0; }

  for (int ct = 0; ct < 128; ++ct) {
    const int colBase = b * NN + ct * 16;

    v2f bf[16];
    load_x_frag(x, colBase + m16, koff, bf);
    const float sqc_self = sq[colBase + m16];

    if (ct + 1 < 128)  // stream next column tile through the cache path
      __builtin_prefetch(x + (size_t)(colBase + 16 + m16) * FIN, 0, 3);

    v8f c0 = {};
    v8f c1 = {};
#pragma unroll
    for (int kc = 0; kc < 16; ++kc) {
      c0 = __builtin_amdgcn_wmma_f32_16x16x4_f32(false, a0[kc], false, bf[kc],
                                                 (short)0, c0, false, false);
      c1 = __builtin_amdgcn_wmma_f32_16x16x4_f32(false, a1[kc], false, bf[kc],
                                                 (short)0, c1, false, false);
    }

#pragma unroll
    for (int r = 0; r < 8; ++r) {
      tile[r + half * 8][m16]      = sqc_self - 2.0f * c0[r];
      tile[16 + r + half * 8][m16] = sqc_self - 2.0f * c1[r];
    }
    __syncthreads();

    // selection: lane l owns row l; scans all 16 columns of this tile
#pragma unroll
    for (int cc = 0; cc < 16; ++cc) {
      const float d = tile[lane][cc] + sqr_self;
      if (d < bd[15]) {
        float dd = d;
        int   ii = ct * 16 + cc;          // event-local neighbor index
#pragma unroll
        for (int t = 0; t < 16; ++t) {
          if (dd < bd[t]) {
            float tf = bd[t]; bd[t] = dd; dd = tf;
            int   ti = bi[t]; bi[t] = ii; ii = ti;
          }
        }
      }
    }
    __syncthreads();
  }

  const int row = rowBase + lane;
#pragma unroll
  for (int j = 0; j < 16; ++j) idx[(size_t)row * KK + j] = bi[j];
}

// ---------------------------------------------------------------------------
// Kernel 3: zero the BN accumulator
// ---------------------------------------------------------------------------
__global__ void __launch_bounds__(128)
init_stats_kernel(float* __restrict__ stats) {
  if (threadIdx.x < 128) stats[threadIdx.x] = 0.f;
}

// ---------------------------------------------------------------------------
// Kernel 4: per-channel sum / sum-of-squares over all 524288 edges.
// Edge value reconstructed as p1[row] + p2[neighbor]  (bias cancels in BN).
// ---------------------------------------------------------------------------
__global__ void __launch_bounds__(256)
stats_kernel(const float* __restrict__ p1, const float* __restrict__ p2,
             const int* __restrict__ idx, float* __restrict__ stats) {
  __shared__ float sd[256];
  __shared__ float sd2[256];
  const int t   = threadIdx.x;
  const int l   = t & 63;
  const int sub = t >> 6;  // 4 edges processed concurrently per block
  const int totalEdges = BB * NN * KK;

  float s = 0.f, s2 = 0.f;
  for (int e = blockIdx.x * 4 + sub; e < totalEdges; e += gridDim.x * 4) {
    const int row  = e >> 4;              // = b*NN + n
    const int b    = row >> 11;           // NN = 2048
    const int gidx = b * NN + idx[e];
    const float v = p1[(size_t)row * LL + l] + p2[(size_t)gidx * LL + l];
    s  += v;
    s2 += v * v;
  }
  sd[t] = s;
  sd2[t] = s2;
  __syncthreads();
  if (t < 64) {
    const float a  = sd[t] + sd[t + 64] + sd[t + 128] + sd[t + 192];
    const float a2 = sd2[t] + sd2[t + 64] + sd2[t + 128] + sd2[t + 192];
    atomicAdd(&stats[t], a);
    atomicAdd(&stats[64 + t], a2);
  }
}

// ---------------------------------------------------------------------------
// Kernel 5: BN(normalize) + ReLU + mean over the K neighbors.
// thread = (row, channel); 4 rows per 256-thread block.
// ---------------------------------------------------------------------------
__global__ void __launch_bounds__(256)
finalize_kernel(const float* __restrict__ p1, const float* __restrict__ p2,
                const int* __restrict__ idx, const float* __restrict__ stats,
                const float* __restrict__ gamma, const float* __restrict__ beta,
                float* __restrict__ out) {
  const int t   = threadIdx.x;
  const int l   = t & 63;
  const int row = blockIdx.x * 4 + (t >> 6);
  const float cnt = (float)(BB * NN * KK);

  const float mean = stats[l] / cnt;
  const float var  = stats[64 + l] / cnt - mean * mean;  // biased var
  const float rs   = rsqrtf(var + EPSBN);
  const float g    = gamma[l];
  const float be   = beta[l];

  const int b = row >> 11;
  const float base = p1[(size_t)row * LL + l];
  float acc = 0.f;
#pragma unroll
  for (int k = 0; k < KK; ++k) {
    const int gi = idx[(size_t)row * KK + k];
    const float v = base + p2[((size_t)(b * NN + gi)) * LL + l];
    const float h = (v - mean) * rs * g + be;
    acc += fmaxf(h, 0.f);
  }
  out[(size_t)row * LL + l] = acc * (1.0f / KK);
}

// ---------------------------------------------------------------------------
extern "C" void kernel_launch(void* const* d_in, const int* in_sizes, int n_in,
                              void* d_out, int out_size, void* d_ws,
                              size_t ws_size, hipStream_t stream) {
  (void)in_sizes; (void)n_in; (void)out_size; (void)ws_size;
  const float* x     = (const float*)d_in[0];
  // d_in[1] = batch (equal-size events encoded structurally) -> unused
  const float* W     = (const float*)d_in[2];
  // d_in[3] = bias: cancels exactly inside BatchNorm (h - mean) -> unused
  const float* gamma = (const float*)d_in[4];
  const float* beta  = (const float*)d_in[5];
  float* out = (float*)d_out;

  char* ws = (char*)d_ws;
  float* sq    = (float*)ws;                                   // 32768 f32
  float* p1    = (float*)(ws + 131072);                        // 8 MB
  float* p2    = (float*)(ws + 131072 + 8388608);              // 8 MB
  int*   idx   = (int*)  (ws + 131072 + 2 * 8388608);          // 2 MB
  float* stats = (float*)(ws + 131072 + 2 * 8388608 + 2097152);// 128 f32

  proj_kernel<<<(BB * NN) / 16, 32, 0, stream>>>(x, W, sq, p1, p2);
  knn_kernel<<<(BB * NN) / 32, 32, 0, stream>>>(x, sq, idx);
  init_stats_kernel<<<1, 128, 0, stream>>>(stats);
  stats_kernel<<<512, 256, 0, stream>>>(p1, p2, idx, stats);
  finalize_kernel<<<(BB * NN) / 4, 256, 0, stream>>>(p1, p2, idx, stats,
                                                     gamma, beta, out);
}